// HGT_35072702939485
// MI455X (gfx1250) — compile-verified
//
#include <hip/hip_runtime.h>
#include <hip/hip_bf16.h>

#define N_NODES 50000
#define N_EDGES 300000
#define HID 128
#define HEADS 4
#define HDIM 32
#define RSQRT_D 0.17677669529663687f   // 1/sqrt(32)
#define LN_EPS 1e-5f
#define BN_EPS 1e-5f

typedef __attribute__((ext_vector_type(2))) float v2f;
typedef __attribute__((ext_vector_type(8))) float v8f;

// ---------------------------------------------------------------------------
// WMMA f32 16x16x4: A 16x4 (2 VGPR), B 4x16 (2 VGPR), C/D 16x16 f32 (8 VGPR)
// ---------------------------------------------------------------------------
__device__ inline v8f wmma4(v2f a, v2f b, v8f c) {
    return __builtin_amdgcn_wmma_f32_16x16x4_f32(
        /*neg_a=*/false, a, /*neg_b=*/false, b,
        /*c_mod=*/(short)0, c, /*reuse_a=*/false, /*reuse_b=*/false);
}

// monotone order-preserving float<->uint for atomicMax-based segment max
__device__ inline unsigned fflip(float f) {
    unsigned u = __float_as_uint(f);
    return (u & 0x80000000u) ? ~u : (u | 0x80000000u);
}
__device__ inline float funflip(unsigned u) {
    u = (u & 0x80000000u) ? (u & 0x7FFFFFFFu) : ~u;
    return __uint_as_float(u);
}

// ---------------------------------------------------------------------------
// Typed GEMM: C[N,OUT] = X[N,F] @ W[ntype[row]][F,OUT] (+ bias[ntype[row]])
// One wave per 16x16 output tile; computes both type-GEMMs, selects per row.
// ---------------------------------------------------------------------------
template <int F, int OUT, bool BIAS>
__global__ void typed_gemm_k(const float* __restrict__ X,
                             const float* __restrict__ W,   // [2,F,OUT]
                             const float* __restrict__ bias,// [2,OUT] or null
                             const int* __restrict__ ntype,
                             float* __restrict__ C, int nTiles) {
    constexpr int TN = OUT / 16;
    int wid  = threadIdx.x >> 5;
    int tile = blockIdx.x * (blockDim.x >> 5) + wid;
    if (tile >= nTiles) return;                 // wave-uniform exit
    int mt = tile / TN, nt = tile % TN;
    int lane = threadIdx.x & 31;
    int half = lane >> 4, lm = lane & 15;
    int rowA = mt * 16 + lm;
    int col  = nt * 16 + lm;
    const float* xr = X + (size_t)rowA * F;
    const float* W0 = W;
    const float* W1 = W + (size_t)F * OUT;
    v8f acc0 = {}; v8f acc1 = {};
#pragma unroll
    for (int k = 0; k < F; k += 4) {
        int k0 = k + 2 * half;                  // lanes 16-31 supply K=2,3
        v2f a = *(const v2f*)(xr + k0);
        v2f b0, b1;
        b0.x = W0[(size_t)k0 * OUT + col];
        b0.y = W0[(size_t)(k0 + 1) * OUT + col];
        b1.x = W1[(size_t)k0 * OUT + col];
        b1.y = W1[(size_t)(k0 + 1) * OUT + col];
        acc0 = wmma4(a, b0, acc0);
        acc1 = wmma4(a, b1, acc1);
    }
    int rbase = mt * 16 + half * 8;             // lanes 16-31 hold rows M=8..15
#pragma unroll
    for (int r = 0; r < 8; ++r) {
        int row = rbase + r;
        int t = ntype[row];
        float v = t ? acc1[r] : acc0[r];
        if (BIAS) v += bias[t * OUT + col];
        C[(size_t)row * OUT + col] = v;
    }
}

// ---------------------------------------------------------------------------
// Per-layer scratch init: agg=0, den=0, amax=fflip(-inf), bnstats=0
// ---------------------------------------------------------------------------
__global__ void init_layer_k(float* __restrict__ agg, float* __restrict__ den,
                             unsigned* __restrict__ amax, float* __restrict__ bnstats) {
    int i = blockIdx.x * blockDim.x + threadIdx.x;
    if (i < N_NODES * HID) agg[i] = 0.f;
    if (i < N_NODES * HEADS) { den[i] = 0.f; amax[i] = 0x007FFFFFu; } // fflip(-inf)
    if (i < 256) bnstats[i] = 0.f;
}

// ---------------------------------------------------------------------------
// Edge pass 1: a[e,h] = (Watt[h,et]^T k_src) . q_dst * pri/sqrt(D); segment max
// ---------------------------------------------------------------------------
__global__ void edge_att_k(const float* __restrict__ K, const float* __restrict__ Q,
                           const int* __restrict__ src, const int* __restrict__ dst,
                           const int* __restrict__ et, const float* __restrict__ Watt,
                           const float* __restrict__ pri, float* __restrict__ a_e,
                           unsigned* __restrict__ amax) {
    int e = blockIdx.x * blockDim.x + threadIdx.x;
    if (e >= N_EDGES) return;
    int s = src[e], d = dst[e], t = et[e];
    const float* kp = K + (size_t)s * HID;
    const float* qp = Q + (size_t)d * HID;
#pragma unroll
    for (int h = 0; h < HEADS; ++h) {
        const float* W = Watt + (size_t)(h * 4 + t) * HDIM * HDIM;
        float qh[HDIM];
#pragma unroll
        for (int o = 0; o < HDIM; ++o) qh[o] = qp[h * HDIM + o];
        float acc = 0.f;
        for (int i = 0; i < HDIM; ++i) {
            float ti = 0.f;
#pragma unroll
            for (int o = 0; o < HDIM; ++o) ti = fmaf(W[i * HDIM + o], qh[o], ti);
            acc = fmaf(kp[h * HDIM + i], ti, acc);
        }
        acc *= pri[h * 4 + t] * RSQRT_D;
        a_e[e * HEADS + h] = acc;
        atomicMax(&amax[d * HEADS + h], fflip(acc));
    }
}

// ---------------------------------------------------------------------------
// Edge pass 2: ex = exp(a - amax[dst]); den += ex; agg[dst] += (Wmsg^T v_src)*ex
// ---------------------------------------------------------------------------
__global__ void edge_msg_k(const float* __restrict__ V, const int* __restrict__ src,
                           const int* __restrict__ dst, const int* __restrict__ et,
                           const float* __restrict__ Wmsg, const float* __restrict__ a_e,
                           const unsigned* __restrict__ amax, float* __restrict__ den,
                           float* __restrict__ agg) {
    int e = blockIdx.x * blockDim.x + threadIdx.x;
    if (e >= N_EDGES) return;
    int s = src[e], d = dst[e], t = et[e];
    const float* vp = V + (size_t)s * HID;
#pragma unroll
    for (int h = 0; h < HEADS; ++h) {
        float am = funflip(amax[d * HEADS + h]);
        float ex = expf(a_e[e * HEADS + h] - am);
        atomicAdd(&den[d * HEADS + h], ex);
        const float* W = Wmsg + (size_t)(h * 4 + t) * HDIM * HDIM;
        float m[HDIM];
#pragma unroll
        for (int o = 0; o < HDIM; ++o) m[o] = 0.f;
        for (int i = 0; i < HDIM; ++i) {
            float vi = vp[h * HDIM + i];
#pragma unroll
            for (int o = 0; o < HDIM; ++o) m[o] = fmaf(vi, W[i * HDIM + o], m[o]);
        }
#pragma unroll
        for (int o = 0; o < HDIM; ++o)
            atomicAdd(&agg[(size_t)d * HID + h * HDIM + o], m[o] * ex);
    }
}

// softmax normalize (0 for nodes with no in-edges)
__global__ void attn_div_k(float* __restrict__ agg, const float* __restrict__ den) {
    int i = blockIdx.x * blockDim.x + threadIdx.x;
    if (i >= N_NODES * HID) return;
    int n = i >> 7;
    int h = (i >> 5) & 3;
    float dd = den[n * HEADS + h];
    agg[i] = dd > 0.f ? agg[i] / dd : 0.f;
}

// ---------------------------------------------------------------------------
// sigmoid-skip + LayerNorm (wave per node) + BN partial-sum accumulation
// ---------------------------------------------------------------------------
__global__ void skip_ln_k(float* __restrict__ H, const float* __restrict__ Xin,
                          int useSkipAdd, const int* __restrict__ ntype,
                          const float* __restrict__ skip, const float* __restrict__ lng,
                          const float* __restrict__ lnb, float* __restrict__ bnstats) {
    __shared__ float sb[256];
    for (int i = threadIdx.x; i < 256; i += blockDim.x) sb[i] = 0.f;
    __syncthreads();
    int wid = threadIdx.x >> 5, lane = threadIdx.x & 31;
    int n = blockIdx.x * 8 + wid;               // N_NODES divisible by 8
    {
        int t = ntype[n];
        float alpha = 1.f / (1.f + expf(-skip[t]));
        float vals[4];
#pragma unroll
        for (int j = 0; j < 4; ++j) {
            int dd = lane + 32 * j;
            float hv = H[(size_t)n * HID + dd] * alpha;
            if (useSkipAdd) hv = fmaf(Xin[(size_t)n * HID + dd], 1.f - alpha, hv);
            vals[j] = hv;
        }
        float s = vals[0] + vals[1] + vals[2] + vals[3];
#pragma unroll
        for (int off = 16; off; off >>= 1) s += __shfl_xor(s, off, 32);
        float mu = s * (1.f / 128.f);
        float vv = 0.f;
#pragma unroll
        for (int j = 0; j < 4; ++j) { float c = vals[j] - mu; vv = fmaf(c, c, vv); }
#pragma unroll
        for (int off = 16; off; off >>= 1) vv += __shfl_xor(vv, off, 32);
        float inv = rsqrtf(vv * (1.f / 128.f) + LN_EPS);
#pragma unroll
        for (int j = 0; j < 4; ++j) {
            int dd = lane + 32 * j;
            float o = fmaf((vals[j] - mu) * inv, lng[dd], lnb[dd]);
            H[(size_t)n * HID + dd] = o;
            atomicAdd(&sb[dd], o);
            atomicAdd(&sb[128 + dd], o * o);
        }
    }
    __syncthreads();
    for (int i = threadIdx.x; i < 256; i += blockDim.x) atomicAdd(&bnstats[i], sb[i]);
}

// BN batch stats -> per-channel affine (scale, shift)
__global__ void bn_finalize_k(const float* __restrict__ bnstats,
                              const float* __restrict__ bng, const float* __restrict__ bnb,
                              float* __restrict__ bnaff) {
    int d = threadIdx.x;
    float mu  = bnstats[d] * (1.f / N_NODES);
    float var = bnstats[128 + d] * (1.f / N_NODES) - mu * mu;
    float sc  = bng[d] * rsqrtf(var + BN_EPS);
    bnaff[d] = sc;
    bnaff[128 + d] = bnb[d] - mu * sc;
}

// ---------------------------------------------------------------------------
// out = (BN(H)) @ lin_W + lin_b   (BN affine fused into A-load), WMMA f32
// ---------------------------------------------------------------------------
__global__ void bn_lin_gemm_k(const float* __restrict__ X, const float* __restrict__ bnsc,
                              const float* __restrict__ bnsh, const float* __restrict__ W,
                              const float* __restrict__ bias, float* __restrict__ C,
                              int nTiles) {
    constexpr int F = HID, OUT = HID, TN = OUT / 16;
    int wid  = threadIdx.x >> 5;
    int tile = blockIdx.x * (blockDim.x >> 5) + wid;
    if (tile >= nTiles) return;
    int mt = tile / TN, nt = tile % TN;
    int lane = threadIdx.x & 31;
    int half = lane >> 4, lm = lane & 15;
    int rowA = mt * 16 + lm;
    int col  = nt * 16 + lm;
    const float* xr = X + (size_t)rowA * F;
    v8f acc = {};
#pragma unroll
    for (int k = 0; k < F; k += 4) {
        int k0 = k + 2 * half;
        v2f a, b;
        a.x = fmaf(xr[k0],     bnsc[k0],     bnsh[k0]);
        a.y = fmaf(xr[k0 + 1], bnsc[k0 + 1], bnsh[k0 + 1]);
        b.x = W[(size_t)k0 * OUT + col];
        b.y = W[(size_t)(k0 + 1) * OUT + col];
        acc = wmma4(a, b, acc);
    }
    int rbase = mt * 16 + half * 8;
#pragma unroll
    for (int r = 0; r < 8; ++r)
        C[(size_t)(rbase + r) * OUT + col] = acc[r] + bias[col];
}

// ---------------------------------------------------------------------------
extern "C" void kernel_launch(void* const* d_in, const int* in_sizes, int n_in,
                              void* d_out, int out_size, void* d_ws, size_t ws_size,
                              hipStream_t stream) {
    (void)in_sizes; (void)n_in; (void)out_size; (void)ws_size;
    // Flattening: top-level insertion order x, params(leaves), src, dst, ntype, etype.
    // params leaves key-sorted: layers[0], layers[1] (each: Wa, Watt, Wk, Wmsg, Wq, Wv,
    // bn_b, bn_g, lin_W, lin_b, ln_b, ln_g, pri, skip), then proj_W, proj_b.
    const float* x = (const float*)d_in[0];
    auto LP = [&](int layer, int off) { return (const float*)d_in[1 + layer * 14 + off]; };
    const float* proj_W = (const float*)d_in[29];
    const float* proj_b = (const float*)d_in[30];
    const int* src   = (const int*)d_in[31];
    const int* dst   = (const int*)d_in[32];
    const int* ntype = (const int*)d_in[33];
    const int* etype = (const int*)d_in[34];

    char* w = (char*)d_ws;
    size_t off = 0;
    auto alloc = [&](size_t bytes) -> char* {
        char* p = w + off; off += (bytes + 255) & ~(size_t)255; return p;
    };
    float*    h0   = (float*)alloc((size_t)N_NODES * 32 * 4);   // proj output
    float*    hmid = (float*)alloc((size_t)N_NODES * HID * 4);  // layer0 out / layer1 in
    float*    Kb   = (float*)alloc((size_t)N_NODES * HID * 4);
    float*    Qb   = (float*)alloc((size_t)N_NODES * HID * 4);
    float*    Vb   = (float*)alloc((size_t)N_NODES * HID * 4);
    float*    agg  = (float*)alloc((size_t)N_NODES * HID * 4);
    float*    a_e  = (float*)alloc((size_t)N_EDGES * HEADS * 4);
    unsigned* amax = (unsigned*)alloc((size_t)N_NODES * HEADS * 4);
    float*    den  = (float*)alloc((size_t)N_NODES * HEADS * 4);
    float*    bnstats = (float*)alloc(256 * 4);
    float*    bnaff   = (float*)alloc(256 * 4);
    float*    H2 = Kb;   // K dead after edge_att; reuse for Wa output

    const int elemBlocks = (N_NODES * HID + 255) / 256;
    const int edgeBlocks = (N_EDGES + 255) / 256;
    const int tiles128 = (N_NODES / 16) * (HID / 16);
    const int gemmBlocks128 = (tiles128 + 7) / 8;

    // Input projection: [N,64] -> [N,32], per-ntype weight + bias
    {
        int tiles = (N_NODES / 16) * (32 / 16);
        typed_gemm_k<64, 32, true><<<(tiles + 7) / 8, 256, 0, stream>>>(
            x, proj_W, proj_b, ntype, h0, tiles);
    }

    const float* layerIn[2]  = { h0, hmid };
    float*       layerOut[2] = { hmid, (float*)d_out };

    for (int L = 0; L < 2; ++L) {
        const float* Wa   = LP(L, 0);
        const float* Watt = LP(L, 1);
        const float* Wk   = LP(L, 2);
        const float* Wmsg = LP(L, 3);
        const float* Wq   = LP(L, 4);
        const float* Wv   = LP(L, 5);
        const float* bn_b = LP(L, 6);
        const float* bn_g = LP(L, 7);
        const float* linW = LP(L, 8);
        const float* linb = LP(L, 9);
        const float* ln_b = LP(L, 10);
        const float* ln_g = LP(L, 11);
        const float* pri  = LP(L, 12);
        const float* skip = LP(L, 13);

        init_layer_k<<<elemBlocks, 256, 0, stream>>>(agg, den, amax, bnstats);

        if (L == 0) {
            typed_gemm_k<32, 128, false><<<gemmBlocks128, 256, 0, stream>>>(
                layerIn[L], Wk, nullptr, ntype, Kb, tiles128);
            typed_gemm_k<32, 128, false><<<gemmBlocks128, 256, 0, stream>>>(
                layerIn[L], Wq, nullptr, ntype, Qb, tiles128);
            typed_gemm_k<32, 128, false><<<gemmBlocks128, 256, 0, stream>>>(
                layerIn[L], Wv, nullptr, ntype, Vb, tiles128);
        } else {
            typed_gemm_k<128, 128, false><<<gemmBlocks128, 256, 0, stream>>>(
                layerIn[L], Wk, nullptr, ntype, Kb, tiles128);
            typed_gemm_k<128, 128, false><<<gemmBlocks128, 256, 0, stream>>>(
                layerIn[L], Wq, nullptr, ntype, Qb, tiles128);
            typed_gemm_k<128, 128, false><<<gemmBlocks128, 256, 0, stream>>>(
                layerIn[L], Wv, nullptr, ntype, Vb, tiles128);
        }

        edge_att_k<<<edgeBlocks, 256, 0, stream>>>(Kb, Qb, src, dst, etype, Watt, pri,
                                                   a_e, amax);
        edge_msg_k<<<edgeBlocks, 256, 0, stream>>>(Vb, src, dst, etype, Wmsg, a_e, amax,
                                                   den, agg);
        attn_div_k<<<elemBlocks, 256, 0, stream>>>(agg, den);

        typed_gemm_k<128, 128, false><<<gemmBlocks128, 256, 0, stream>>>(
            agg, Wa, nullptr, ntype, H2, tiles128);

        skip_ln_k<<<N_NODES / 8, 256, 0, stream>>>(H2, layerIn[L], (L == 1) ? 1 : 0,
                                                   ntype, skip, ln_g, ln_b, bnstats);
        bn_finalize_k<<<1, 128, 0, stream>>>(bnstats, bn_g, bn_b, bnaff);
        bn_lin_gemm_k<<<gemmBlocks128, 256, 0, stream>>>(H2, bnaff, bnaff + 128, linW,
                                                         linb, layerOut[L], tiles128);
    }
}